// Regression_Cubical_Network_One_Nonlinearity_38946763440762
// MI455X (gfx1250) — compile-verified
//
#include <hip/hip_runtime.h>

typedef float v2f __attribute__((ext_vector_type(2)));
typedef float v8f __attribute__((ext_vector_type(8)));

#define BATCH   16384
#define DIMS    8
#define NHID    4096
#define NLAB    64
#define KC      128              // K-chunk staged in LDS; divides group width 512
#define PITCH   (KC + 4)         // float pitch; %4==0 keeps 16B alignment, staggers banks
#define ROWS_PER_BLOCK 128
#define THREADS 256

__global__ __launch_bounds__(THREADS)
void cubical_fused_wmma(const float* __restrict__ x,
                        const float* __restrict__ swm,
                        const float* __restrict__ coeff,
                        const float* __restrict__ bias,
                        const float* __restrict__ W2,
                        const float* __restrict__ b2,
                        const float* __restrict__ Wout,
                        const float* __restrict__ bout,
                        float* __restrict__ out)
{
    __shared__ float s_swm[DIMS * DIMS];
    __shared__ float s_coeff[KC];
    __shared__ float s_bias[KC];
    __shared__ __align__(16) float s_w2[NLAB * PITCH];   // ~34 KB

    const int t    = threadIdx.x;
    const int lane = t & 31;            // wave32
    const int wave = t >> 5;
    const int half = lane >> 4;         // 0: lanes 0-15 (K+0,1), 1: lanes 16-31 (K+2,3)
    const int sub  = lane & 15;
    const int rowbase = blockIdx.x * ROWS_PER_BLOCK + wave * 16;
    const int myrow   = rowbase + sub;  // batch row this lane feeds into the A fragment

    if (t < DIMS * DIMS) s_swm[t] = swm[t];
    __syncthreads();

    // proj[g] = sum_i x[myrow,i] * swm[g,i]   (tiny 8x8, VALU)
    float xr[DIMS];
#pragma unroll
    for (int i = 0; i < DIMS; ++i) xr[i] = x[myrow * DIMS + i];
    float proj[DIMS];
#pragma unroll
    for (int g = 0; g < DIMS; ++g) {
        float a = 0.f;
#pragma unroll
        for (int i = 0; i < DIMS; ++i) a = __builtin_fmaf(xr[i], s_swm[g * DIMS + i], a);
        proj[g] = a;
    }

    v8f acc0 = {}, acc1 = {}, acc2 = {}, acc3 = {};   // label tiles n0 = 0,16,32,48

    for (int k0 = 0; k0 < NHID; k0 += KC) {
        // ---- stage coeff/bias chunk ----
        if (t < KC)            s_coeff[t]      = coeff[k0 + t];
        else                   s_bias[t - KC]  = bias[k0 + (t - KC)];

        // ---- stage W2 chunk (64 x KC floats) as float4, coalesced along K ----
#pragma unroll
        for (int p = 0; p < (NLAB * KC / 4) / THREADS; ++p) {
            const int idx = p * THREADS + t;       // float4 index
            const int r   = idx >> 5;              // KC/4 == 32 float4 per row
            const int c4  = idx & 31;
            const float4 v = *reinterpret_cast<const float4*>(&W2[r * NHID + k0 + c4 * 4]);
            *reinterpret_cast<float4*>(&s_w2[r * PITCH + c4 * 4]) = v;
            if (k0 + KC < NHID)
                __builtin_prefetch(&W2[r * NHID + k0 + KC + c4 * 4], 0, 1);
        }
        __syncthreads();

        // group id is constant across the chunk (KC divides 512)
        const float pv = proj[k0 >> 9];

#pragma unroll 4
        for (int kk = 0; kk < KC; kk += 4) {
            const int kb = kk + half * 2;
            // A fragment: act values built in registers (fma + clamp to [0,1])
            const float c0 = s_coeff[kb],     bb0 = s_bias[kb];
            const float c1 = s_coeff[kb + 1], bb1 = s_bias[kb + 1];
            v2f a;
            a.x = fminf(fmaxf(__builtin_fmaf(pv, c0, bb0), 0.f), 1.f);
            a.y = fminf(fmaxf(__builtin_fmaf(pv, c1, bb1), 0.f), 1.f);

            // B fragments: W2^T tiles from LDS (8B-aligned pair -> ds_load_b64)
            const v2f bf0 = *reinterpret_cast<const v2f*>(&s_w2[(0 * 16 + sub) * PITCH + kb]);
            const v2f bf1 = *reinterpret_cast<const v2f*>(&s_w2[(1 * 16 + sub) * PITCH + kb]);
            const v2f bf2 = *reinterpret_cast<const v2f*>(&s_w2[(2 * 16 + sub) * PITCH + kb]);
            const v2f bf3 = *reinterpret_cast<const v2f*>(&s_w2[(3 * 16 + sub) * PITCH + kb]);

            acc0 = __builtin_amdgcn_wmma_f32_16x16x4_f32(false, a, false, bf0, (short)0, acc0, false, false);
            acc1 = __builtin_amdgcn_wmma_f32_16x16x4_f32(false, a, false, bf1, (short)0, acc1, false, false);
            acc2 = __builtin_amdgcn_wmma_f32_16x16x4_f32(false, a, false, bf2, (short)0, acc2, false, false);
            acc3 = __builtin_amdgcn_wmma_f32_16x16x4_f32(false, a, false, bf3, (short)0, acc3, false, false);
        }
        __syncthreads();
    }

    // ---- epilogue: out[b] = sum_l labels[b,l]*Wout[l] + (sum_l b2[l]*Wout[l] + bout) ----
    const float w0 = Wout[0 * 16 + sub];
    const float w1 = Wout[1 * 16 + sub];
    const float w2 = Wout[2 * 16 + sub];
    const float w3 = Wout[3 * 16 + sub];

    float s[8];
#pragma unroll
    for (int v = 0; v < 8; ++v)
        s[v] = acc0[v] * w0 + acc1[v] * w1 + acc2[v] * w2 + acc3[v] * w3;

    // reduce across the 16 lanes of each half-wave (rows M=v stay in VGPR v)
#pragma unroll
    for (int v = 0; v < 8; ++v) {
        s[v] += __shfl_xor(s[v], 1, 32);
        s[v] += __shfl_xor(s[v], 2, 32);
        s[v] += __shfl_xor(s[v], 4, 32);
        s[v] += __shfl_xor(s[v], 8, 32);
    }

    float cterm = bout[0];
    for (int l = 0; l < NLAB; ++l) cterm = __builtin_fmaf(b2[l], Wout[l], cterm);

    if (sub == 0) {
        // lane 0 holds rows M=0..7 of this wave's strip, lane 16 holds M=8..15
#pragma unroll
        for (int v = 0; v < 8; ++v)
            out[rowbase + half * 8 + v] = s[v] + cterm;
    }
}

extern "C" void kernel_launch(void* const* d_in, const int* in_sizes, int n_in,
                              void* d_out, int out_size, void* d_ws, size_t ws_size,
                              hipStream_t stream) {
    const float* x     = (const float*)d_in[0];
    const float* swm   = (const float*)d_in[1];
    const float* coeff = (const float*)d_in[2];
    const float* bias  = (const float*)d_in[3];
    const float* W2    = (const float*)d_in[4];
    const float* b2    = (const float*)d_in[5];
    const float* Wout  = (const float*)d_in[6];
    const float* bout  = (const float*)d_in[7];
    float* out = (float*)d_out;

    dim3 grid(BATCH / ROWS_PER_BLOCK);   // 128 blocks
    dim3 block(THREADS);                 // 8 waves
    cubical_fused_wmma<<<grid, block, 0, stream>>>(x, swm, coeff, bias, W2, b2, Wout, bout, out);
}